// TransformerDecoder_65652870087463
// MI455X (gfx1250) — compile-verified
//
#include <hip/hip_runtime.h>
#include <hip/hip_bf16.h>

typedef __attribute__((ext_vector_type(2))) float v2f;
typedef __attribute__((ext_vector_type(8))) float v8f;

#define E 256
#define SEQ_L 512
#define M_TOT 32768           // B*L = 64*512
#define BM 64                 // rows per block
#define BN 128                // cols per block
#define LSTRIDE 260           // padded LDS row stride (floats): 260 % 64 == 4 -> conflict-free b64 reads
#define NLAYER 4

// ---------------------------------------------------------------------------
// Repack conv_w [NL][E_out][E_in][3] -> Wt [NL][2][E/2][E_out][2]
//   Wt[((l*2+t)*(E/2) + e2)*E*2 + o*2 + j] = conv_w[((l*E+o)*E + 2*e2+j)*3 + t]
// Tap 2 is dropped: tri[i,i+1] == 0 always masks it in the reference.
// ---------------------------------------------------------------------------
__global__ __launch_bounds__(256) void wtrans_kernel(const float* __restrict__ cw,
                                                     float* __restrict__ wt) {
  int idx = blockIdx.x * 256 + threadIdx.x;      // 4*2*128*256*2 = 524288 total
  int j  = idx & 1;
  int o  = (idx >> 1) & 255;
  int e2 = (idx >> 9) & 127;
  int t  = (idx >> 16) & 1;
  int l  = idx >> 17;
  int e  = e2 * 2 + j;
  wt[idx] = cw[(((size_t)l * E + o) * E + e) * 3 + t];
}

// ---------------------------------------------------------------------------
// One conv layer: Xout[m,o] = sum_e W0[o,e]*Xin[m-1,e]*(m%512!=0) + W1[o,e]*Xin[m,e]
// Block: 256 threads = 8 waves arranged 4(M) x 2(N). Wave tile: 16 x 64.
// ---------------------------------------------------------------------------
__global__ __launch_bounds__(256) void conv_layer_kernel(const float* __restrict__ Xin,
                                                         const float* __restrict__ Wt,
                                                         float* __restrict__ Xout) {
  __shared__ float ldsX[(BM + 1) * LSTRIDE];

  const int tid  = threadIdx.x;
  const int lane = tid & 31;
  const int wave = tid >> 5;
  const int mb   = blockIdx.x * BM;
  const int nb   = blockIdx.y * BN;

  // Stage X rows [mb-1, mb+BM) into LDS rows [0, BM]. LDS row r = X[mb-1+r].
  // Row 0 is zero when mb is a batch start (covers mb==0 too).
  const bool bstart = (mb % SEQ_L) == 0;
  for (int idx = tid; idx < (BM + 1) * (E / 4); idx += 256) {
    const int r  = idx >> 6;            // E/4 == 64 float4 per row
    const int c4 = idx & 63;
    float4 v;
    if (r == 0 && bstart) {
      v = make_float4(0.f, 0.f, 0.f, 0.f);
    } else {
      v = ((const float4*)(Xin + (size_t)(mb - 1 + r) * E))[c4];
    }
    float* dst = ldsX + r * LSTRIDE + c4 * 4;
    dst[0] = v.x; dst[1] = v.y; dst[2] = v.z; dst[3] = v.w;
  }
  __syncthreads();

  const int wm    = wave >> 1;          // 0..3 : 16-row strip
  const int wn    = wave & 1;           // 0..1 : 64-col strip
  const int r0    = wm * 16;
  const int kh    = (lane >> 4) << 1;   // lanes 16..31 carry K+2 (ISA A/B f32 layout)
  const int nlo   = lane & 15;
  const int ncol0 = nb + wn * 64 + nlo;

  v8f acc0 = {}, acc1 = {}, acc2 = {}, acc3 = {};

#pragma unroll
  for (int t = 0; t < 2; ++t) {
    // tap 0 -> A row = LDS row (r0 + m + 0) == X[m-1]; tap 1 -> +1 == X[m]
    const float* __restrict__ arow = ldsX + (size_t)(r0 + nlo + t) * LSTRIDE;
    const v2f*   __restrict__ Wtt  = (const v2f*)(Wt + (size_t)t * E * E);
    for (int k0 = 0; k0 < E; k0 += 4) {
      const int e = k0 + kh;
      v2f a = *(const v2f*)(arow + e);                  // ds_load_b64, bank-conflict-free
      const v2f* bp = Wtt + ((size_t)(e >> 1) * E + ncol0);
      v2f b0 = bp[0], b1 = bp[16], b2 = bp[32], b3 = bp[48];
      acc0 = __builtin_amdgcn_wmma_f32_16x16x4_f32(false, a, false, b0, (short)0, acc0, false, false);
      acc1 = __builtin_amdgcn_wmma_f32_16x16x4_f32(false, a, false, b1, (short)0, acc1, false, false);
      acc2 = __builtin_amdgcn_wmma_f32_16x16x4_f32(false, a, false, b2, (short)0, acc2, false, false);
      acc3 = __builtin_amdgcn_wmma_f32_16x16x4_f32(false, a, false, b3, (short)0, acc3, false, false);
    }
  }

  // C/D layout: VGPR v holds M = v + 8*(lane>=16), N = lane&15.
  const int mrow = mb + r0 + ((lane >> 4) << 3);
  float* outp = Xout + (size_t)mrow * E + ncol0;
#pragma unroll
  for (int v = 0; v < 8; ++v) {
    outp[(size_t)v * E +  0] = acc0[v];
    outp[(size_t)v * E + 16] = acc1[v];
    outp[(size_t)v * E + 32] = acc2[v];
    outp[(size_t)v * E + 48] = acc3[v];
  }
}

// ---------------------------------------------------------------------------
// fc_out: out[m] = dot(X[m,:], fc_w) + fc_b[0].  One wave per row.
// ---------------------------------------------------------------------------
__global__ __launch_bounds__(256) void fc_kernel(const float* __restrict__ X,
                                                 const float* __restrict__ w,
                                                 const float* __restrict__ b,
                                                 float* __restrict__ out) {
  const int wave = threadIdx.x >> 5;
  const int lane = threadIdx.x & 31;
  const int m = blockIdx.x * 8 + wave;
  const float* xr = X + (size_t)m * E;
  float s = 0.f;
#pragma unroll
  for (int e = lane; e < E; e += 32) s += xr[e] * w[e];
#pragma unroll
  for (int off = 16; off > 0; off >>= 1) s += __shfl_down(s, off);
  if (lane == 0) out[m] = s + b[0];
}

// ---------------------------------------------------------------------------
extern "C" void kernel_launch(void* const* d_in, const int* in_sizes, int n_in,
                              void* d_out, int out_size, void* d_ws, size_t ws_size,
                              hipStream_t stream) {
  const float* x      = (const float*)d_in[0];   // [64,512,256]
  const float* conv_w = (const float*)d_in[1];   // [4,256,256,3]
  const float* fc_w   = (const float*)d_in[2];   // [1,256]
  const float* fc_b   = (const float*)d_in[3];   // [1]
  float* out = (float*)d_out;                    // [64,1,512] -> flat m

  float* ws   = (float*)d_ws;
  float* Wt   = ws;                              // 4*2*256*256 floats = 2 MB
  float* bufA = Wt + (size_t)NLAYER * 2 * E * E; // 32 MB
  float* bufB = bufA + (size_t)M_TOT * E;        // 32 MB

  wtrans_kernel<<<(NLAYER * 2 * E * E) / 256, 256, 0, stream>>>(conv_w, Wt);

  dim3 grid(M_TOT / BM, E / BN);
  const size_t WL = (size_t)2 * E * E;           // floats per layer in Wt
  conv_layer_kernel<<<grid, 256, 0, stream>>>(x,    Wt + 0 * WL, bufA);
  conv_layer_kernel<<<grid, 256, 0, stream>>>(bufA, Wt + 1 * WL, bufB);
  conv_layer_kernel<<<grid, 256, 0, stream>>>(bufB, Wt + 2 * WL, bufA);
  conv_layer_kernel<<<grid, 256, 0, stream>>>(bufA, Wt + 3 * WL, bufB);

  fc_kernel<<<M_TOT / 8, 256, 0, stream>>>(bufB, fc_w, fc_b, out);
}